// Encoder_42279658062345
// MI455X (gfx1250) — compile-verified
//
#include <hip/hip_runtime.h>
#include <hip/hip_bf16.h>

// ---------------------------------------------------------------------------
// 2-layer GCN (PyG GCNConv semantics) for MI455X / gfx1250, fp32 end-to-end.
//   layer: out = relu( scatter_add( (x@W)[src] * dinv[src]*dinv[dst] -> dst )
//                      + (x@W)*dinv^2 (self loop) + b )
// GEMMs use V_WMMA_F32_16X16X4_F32 (fp32 matrix pipe, matches ref precision).
// Scatter uses float global atomics (global_atomic_add_f32).
// ---------------------------------------------------------------------------

typedef __attribute__((ext_vector_type(2))) float v2f;
typedef __attribute__((ext_vector_type(8))) float v8f;

// ---------------- degree / normalization -----------------------------------

__global__ void k_init_deg(float* __restrict__ deg, int n) {
    int i = blockIdx.x * blockDim.x + threadIdx.x;
    if (i < n) deg[i] = 1.0f;                     // self-loop contributes 1
}

__global__ void k_count_deg(float* __restrict__ deg, const int* __restrict__ ei,
                            int nEdges) {
    int e = blockIdx.x * blockDim.x + threadIdx.x;
    if (e < nEdges) {
        int t = ei[(size_t)nEdges + e];           // dst row of edge_index
        unsafeAtomicAdd(&deg[t], 1.0f);
    }
}

__global__ void k_rsqrt_inplace(float* __restrict__ d, int n) {
    int i = blockIdx.x * blockDim.x + threadIdx.x;
    if (i < n) d[i] = __frsqrt_rn(d[i]);          // deg >= 1 always (self loop)
}

// ---------------- fp32 WMMA GEMM:  C[nRows,nCols] = A[nRows,K] @ B[K,nCols] --
// Wave computes a 16x64 strip via 4 accumulators; 8 waves/block -> 128x64 tile.
// A-frag (16x4 f32, 2 VGPR): lane l holds M=l&15, K = 2*(l>>4) + {0,1}
// B-frag (4x16 f32, 2 VGPR): lane l holds N=l&15, K = 2*(l>>4) + {0,1}
// C/D  (16x16 f32, 8 VGPR): vgpr i, lanes 0-15 -> M=i, lanes 16-31 -> M=i+8
// RELU is a compile-time flag (fuses layer-1 activation into layer-2 A-load
// without per-iteration cndmask in the inner loop).

template <bool RELU>
__global__ __launch_bounds__(256)
void gcn_gemm_wmma_f32(const float* __restrict__ A, const float* __restrict__ Bm,
                       float* __restrict__ Cm, int nRows, int K, int nCols) {
    const int lane = threadIdx.x & 31;
    const int wv   = threadIdx.x >> 5;
    const int rowBase = blockIdx.x * 128 + wv * 16;
    const int colBase = blockIdx.y * 64;
    const int m  = lane & 15;
    const int lr = lane >> 4;          // lane half: 0 or 1
    const int kg = lr * 2;             // per-lane K base inside a 4-wide step

    int arow = rowBase + m;            // clamp OOB rows (stores are predicated)
    if (arow > nRows - 1) arow = nRows - 1;
    const float* Ar = A + (size_t)arow * K;

    v8f c0 = {}, c1 = {}, c2 = {}, c3 = {};

    for (int k0 = 0; k0 < K; k0 += 4) {
        const int ka = k0 + kg;
        v2f a;
        a.x = Ar[ka];
        a.y = Ar[ka + 1];
        if (RELU) { a.x = fmaxf(a.x, 0.0f); a.y = fmaxf(a.y, 0.0f); }

        const float* B0 = Bm + (size_t)ka * nCols + colBase + m;  // row ka
        const float* B1 = B0 + nCols;                             // row ka+1
        v2f b0; b0.x = B0[0];  b0.y = B1[0];
        v2f b1; b1.x = B0[16]; b1.y = B1[16];
        v2f b2; b2.x = B0[32]; b2.y = B1[32];
        v2f b3; b3.x = B0[48]; b3.y = B1[48];

        c0 = __builtin_amdgcn_wmma_f32_16x16x4_f32(false, a, false, b0, (short)0, c0, false, false);
        c1 = __builtin_amdgcn_wmma_f32_16x16x4_f32(false, a, false, b1, (short)0, c1, false, false);
        c2 = __builtin_amdgcn_wmma_f32_16x16x4_f32(false, a, false, b2, (short)0, c2, false, false);
        c3 = __builtin_amdgcn_wmma_f32_16x16x4_f32(false, a, false, b3, (short)0, c3, false, false);
    }

#pragma unroll
    for (int i = 0; i < 8; ++i) {
        const int row = rowBase + i + 8 * lr;
        if (row < nRows) {
            float* Cp = Cm + (size_t)row * nCols + colBase + m;
            Cp[0]  = c0[i];
            Cp[16] = c1[i];
            Cp[32] = c2[i];
            Cp[48] = c3[i];
        }
    }
}

// ---------------- self-loop init:  out = h * dinv_i^2 + bias ----------------
// float4-vectorized (d is a multiple of 4).

__global__ void k_self_init(float4* __restrict__ out, const float4* __restrict__ h,
                            const float* __restrict__ dinv,
                            const float4* __restrict__ bias, int nNodes, int d4) {
    size_t gid = (size_t)blockIdx.x * blockDim.x + threadIdx.x;
    size_t total = (size_t)nNodes * (size_t)d4;
    if (gid >= total) return;
    int i = (int)(gid / d4);
    int c = (int)(gid % d4);
    float di = dinv[i];
    float s = di * di;
    float4 hv = h[gid];
    float4 bv = bias[c];
    float4 o;
    o.x = hv.x * s + bv.x;
    o.y = hv.y * s + bv.y;
    o.z = hv.z * s + bv.z;
    o.w = hv.w * s + bv.w;
    out[gid] = o;
}

// ---------------- edge scatter: out[dst] += h[src] * dinv[src]*dinv[dst] ----
// d/4 lanes cooperate per edge; each lane moves one float4 (coalesced gather,
// 4x global_atomic_add_f32 scatter).

__global__ void k_scatter(float* __restrict__ out, const float* __restrict__ h,
                          const float* __restrict__ dinv,
                          const int* __restrict__ ei, int nEdges, int d) {
    const int lpe = d >> 2;                       // lanes per edge
    size_t gid = (size_t)blockIdx.x * blockDim.x + threadIdx.x;
    size_t e = gid / (size_t)lpe;
    int q = (int)(gid % (size_t)lpe);
    if (e >= (size_t)nEdges) return;

    int s = ei[e];
    int t = ei[(size_t)nEdges + e];
    float nrm = dinv[s] * dinv[t];

    const float4* hp = (const float4*)(h + (size_t)s * d);
    float4 v = hp[q];
    float* op = out + (size_t)t * d + (size_t)q * 4;
    unsafeAtomicAdd(op + 0, v.x * nrm);
    unsafeAtomicAdd(op + 1, v.y * nrm);
    unsafeAtomicAdd(op + 2, v.z * nrm);
    unsafeAtomicAdd(op + 3, v.w * nrm);
}

// ---------------- final ReLU (float4-vectorized) ----------------------------

__global__ void k_relu4(float4* __restrict__ p, size_t n4) {
    size_t gid = (size_t)blockIdx.x * blockDim.x + threadIdx.x;
    if (gid < n4) {
        float4 v = p[gid];
        v.x = fmaxf(v.x, 0.0f);
        v.y = fmaxf(v.y, 0.0f);
        v.z = fmaxf(v.z, 0.0f);
        v.w = fmaxf(v.w, 0.0f);
        p[gid] = v;
    }
}

// ---------------------------------------------------------------------------

extern "C" void kernel_launch(void* const* d_in, const int* in_sizes, int n_in,
                              void* d_out, int out_size, void* d_ws, size_t ws_size,
                              hipStream_t stream) {
    const float* x  = (const float*)d_in[0];
    const int*   ei = (const int*)d_in[1];   // [2,E] flat: row0=src, row1=dst
    const float* W1 = (const float*)d_in[2];
    const float* b1 = (const float*)d_in[3];
    const float* W2 = (const float*)d_in[4];
    const float* b2 = (const float*)d_in[5];
    float* out = (float*)d_out;

    const int DIN = 128, DH = 256, DOUT = 128;
    const int N = in_sizes[0] / DIN;
    const int E = in_sizes[1] / 2;

    float* ws   = (float*)d_ws;
    float* dinv = ws;                        // N floats
    float* bufA = ws + N;                    // N*DH floats (h1, reused for h2)
    float* bufB = bufA + (size_t)N * DH;     // N*DH floats (layer-1 pre-relu out)

    dim3 blk(256);
    auto cdiv = [](size_t a, size_t b) { return (unsigned)((a + b - 1) / b); };

    // --- degree / dinv (shared by both layers) ---
    k_init_deg<<<cdiv(N, 256), blk, 0, stream>>>(dinv, N);
    k_count_deg<<<cdiv(E, 256), blk, 0, stream>>>(dinv, ei, E);
    k_rsqrt_inplace<<<cdiv(N, 256), blk, 0, stream>>>(dinv, N);

    // --- layer 1: h1 = x @ W1 ; agg1 = scatter(h1) + self + b1 (pre-relu) ---
    gcn_gemm_wmma_f32<false><<<dim3(cdiv(N, 128), DH / 64), blk, 0, stream>>>(
        x, W1, bufA, N, DIN, DH);
    k_self_init<<<cdiv((size_t)N * (DH / 4), 256), blk, 0, stream>>>(
        (float4*)bufB, (const float4*)bufA, dinv, (const float4*)b1, N, DH / 4);
    k_scatter<<<cdiv((size_t)E * (DH / 4), 256), blk, 0, stream>>>(bufB, bufA, dinv, ei, E, DH);

    // --- layer 2: h2 = relu(agg1) @ W2 (ReLU fused into A-load) ---
    gcn_gemm_wmma_f32<true><<<dim3(cdiv(N, 128), DOUT / 64), blk, 0, stream>>>(
        bufB, W2, bufA, N, DH, DOUT);
    k_self_init<<<cdiv((size_t)N * (DOUT / 4), 256), blk, 0, stream>>>(
        (float4*)out, (const float4*)bufA, dinv, (const float4*)b2, N, DOUT / 4);
    k_scatter<<<cdiv((size_t)E * (DOUT / 4), 256), blk, 0, stream>>>(out, bufA, dinv, ei, E, DOUT);

    // --- final ReLU ---
    k_relu4<<<cdiv((size_t)N * (DOUT / 4), 256), blk, 0, stream>>>(
        (float4*)out, (size_t)N * (DOUT / 4));
}